// Transformer_28183575396373
// MI455X (gfx1250) — compile-verified
//
#include <hip/hip_runtime.h>
#include <math.h>
#include <stdint.h>

// ---------------- constants ----------------
#define DIMC   1024
#define NSEQ   1024
#define BSZ    2
#define NHEAD  16
#define DH     64
#define NLAYER 6
#define DFFC   2730
#define DCOND  4096
#define MROWS  (BSZ * NSEQ)        // 2048
#define NEGMAX (-3.4028234663852886e38f)

typedef __attribute__((ext_vector_type(2))) float v2f;
typedef __attribute__((ext_vector_type(8))) float v8f;

static __device__ __forceinline__ v8f wmma4(v2f a, v2f b, v8f c) {
  // D(16x16,f32) = A(16x4,f32) * B(4x16,f32) + C
  return __builtin_amdgcn_wmma_f32_16x16x4_f32(false, a, false, b, (short)0, c,
                                               false, false);
}

static __device__ __forceinline__ float sigmoidf_(float x) {
  return 1.0f / (1.0f + expf(-x));
}

// async Global->LDS helpers (CDNA5 GLOBAL_LOAD_ASYNC_TO_LDS_*, ASYNCcnt)
static __device__ __forceinline__ void async_b128(unsigned lds, const float* g) {
  asm volatile("global_load_async_to_lds_b128 %0, %1, off" ::"v"(lds), "v"(g)
               : "memory");
}
static __device__ __forceinline__ void async_b32(unsigned lds, const float* g) {
  asm volatile("global_load_async_to_lds_b32 %0, %1, off" ::"v"(lds), "v"(g)
               : "memory");
}
static __device__ __forceinline__ void wait_async0() {
  asm volatile("s_wait_asynccnt 0" ::: "memory");
}
static __device__ __forceinline__ unsigned lds_addr(const void* p) {
  return (unsigned)(uintptr_t)p;  // low 32 bits of generic ptr = LDS offset
}

// ---------------- generic fp32 WMMA GEMM: C = A[MxK] @ W[KxN] (+bias) --------
__global__ __launch_bounds__(256) void k_gemm(const float* __restrict__ A,
                                              const float* __restrict__ W,
                                              const float* __restrict__ bias,
                                              float* __restrict__ C,
                                              int M, int K, int N) {
  __shared__ float As[64][20];   // stride 20: 16B-aligned rows, conflict-free
  __shared__ float Bs[16][64];   // read row-consecutive: conflict-free
  const int tid  = threadIdx.x;
  const int wave = tid >> 5;
  const int lane = tid & 31;
  const int hf   = lane >> 4;   // half-wave
  const int lr   = lane & 15;
  const int bm   = blockIdx.y * 64;
  const int bn   = blockIdx.x * 64;
  const int m0   = (wave & 3) * 16;
  const int n0a  = (wave >> 2) * 32;
  const int n0b  = n0a + 16;

  v8f acc0 = {0.f,0.f,0.f,0.f,0.f,0.f,0.f,0.f};
  v8f acc1 = {0.f,0.f,0.f,0.f,0.f,0.f,0.f,0.f};

  for (int k0 = 0; k0 < K; k0 += 16) {
    const bool interior =
        (bm + 64 <= M) && (k0 + 16 <= K) && (bn + 64 <= N) && ((N & 3) == 0);
    if (interior) {
      {  // A tile 64x16: one b128 (or 4x b32 when rows unaligned) per thread
        const int r = tid >> 2;
        const int c4 = (tid & 3) * 4;
        const float* ga = A + (long)(bm + r) * K + (k0 + c4);
        unsigned la = lds_addr(&As[r][c4]);
        if ((K & 3) == 0) {
          async_b128(la, ga);
        } else {
          async_b32(la, ga);
          async_b32(la + 4, ga + 1);
          async_b32(la + 8, ga + 2);
          async_b32(la + 12, ga + 3);
        }
      }
      {  // B tile 16x64: one b128 per thread
        const int r = tid >> 4;
        const int c4 = (tid & 15) * 4;
        const float* gb = W + (long)(k0 + r) * N + (bn + c4);
        async_b128(lds_addr(&Bs[r][c4]), gb);
      }
      wait_async0();
    } else {
      for (int idx = tid; idx < 64 * 16; idx += 256) {
        int r = idx >> 4, c = idx & 15;
        int gr = bm + r, gc = k0 + c;
        As[r][c] = (gr < M && gc < K) ? A[(long)gr * K + gc] : 0.0f;
      }
      for (int idx = tid; idx < 16 * 64; idx += 256) {
        int r = idx >> 6, c = idx & 63;
        int gr = k0 + r, gc = bn + c;
        Bs[r][c] = (gr < K && gc < N) ? W[(long)gr * N + gc] : 0.0f;
      }
    }
    __syncthreads();
#pragma unroll
    for (int kk = 0; kk < 16; kk += 4) {
      int ks = kk + hf * 2;
      v2f a  = {As[m0 + lr][ks], As[m0 + lr][ks + 1]};
      v2f b0 = {Bs[ks][n0a + lr], Bs[ks + 1][n0a + lr]};
      v2f b1 = {Bs[ks][n0b + lr], Bs[ks + 1][n0b + lr]};
      acc0 = wmma4(a, b0, acc0);
      acc1 = wmma4(a, b1, acc1);
    }
    __syncthreads();
  }
#pragma unroll
  for (int r = 0; r < 8; ++r) {
    int row = bm + m0 + hf * 8 + r;
    if (row < M) {
      int c0 = bn + n0a + lr;
      int c1 = bn + n0b + lr;
      if (c0 < N) C[(long)row * N + c0] = acc0[r] + (bias ? bias[c0] : 0.0f);
      if (c1 < N) C[(long)row * N + c1] = acc1[r] + (bias ? bias[c1] : 0.0f);
    }
  }
}

// ---------------- time conditioning: cond = silu(fourier @ W + b) ------------
__global__ __launch_bounds__(256) void k_cond(const float* __restrict__ times,
                                              const float* __restrict__ fw,
                                              const float* __restrict__ tw,
                                              const float* __restrict__ tb,
                                              float* __restrict__ cond) {
  __shared__ float fs[1025];
  const int b = blockIdx.y;
  float t = times[b];
  for (int j = threadIdx.x; j < 512; j += 256) {
    float f = t * fw[j] * 6.283185307179586f;
    fs[1 + j]       = sinf(f);
    fs[1 + 512 + j] = cosf(f);
  }
  if (threadIdx.x == 0) fs[0] = t;
  __syncthreads();
  int col = blockIdx.x * 256 + threadIdx.x;  // 0..4095
  float acc = tb[col];
  for (int k = 0; k < 1025; ++k) acc += fs[k] * tw[(long)k * DCOND + col];
  cond[b * DCOND + col] = acc * sigmoidf_(acc);  // silu
}

// ---------------- per-layer FiLM / adaZero vectors ---------------------------
// films layout per (l,b): [0:1024) attn_g, [1024:2048) attn_b,
// [2048:3072) attn_adaz(sig), [3072:4096) ff_g, [4096:5120) ff_b,
// [5120:6144) ff_adaz(sig)
__global__ __launch_bounds__(256) void k_films(
    const float* __restrict__ cond, const float* __restrict__ afw,
    const float* __restrict__ afb, const float* __restrict__ azw,
    const float* __restrict__ azb, const float* __restrict__ ffw,
    const float* __restrict__ ffb, const float* __restrict__ fzw,
    const float* __restrict__ fzb, float* __restrict__ films) {
  __shared__ float cs[DCOND];
  const int b = blockIdx.y, l = blockIdx.z;
  for (int i = threadIdx.x; i < DCOND; i += 256) cs[i] = cond[b * DCOND + i];
  __syncthreads();
  int slot = blockIdx.x * 256 + threadIdx.x;  // 0..6143
  const float* w;
  long stride;
  float acc;
  int dosig = 0;
  if (slot < 2048) {
    w = afw + (long)l * DCOND * 2048 + slot; stride = 2048;
    acc = afb[l * 2048 + slot];
  } else if (slot < 3072) {
    int c = slot - 2048;
    w = azw + (long)l * DCOND * 1024 + c; stride = 1024;
    acc = azb[l * 1024 + c]; dosig = 1;
  } else if (slot < 5120) {
    int c = slot - 3072;
    w = ffw + (long)l * DCOND * 2048 + c; stride = 2048;
    acc = ffb[l * 2048 + c];
  } else {
    int c = slot - 5120;
    w = fzw + (long)l * DCOND * 1024 + c; stride = 1024;
    acc = fzb[l * 1024 + c]; dosig = 1;
  }
  for (int k = 0; k < DCOND; ++k) acc += cs[k] * w[(long)k * stride];
  if (dosig) acc = sigmoidf_(acc);
  films[((long)l * BSZ + b) * 6144 + slot] = acc;
}

// ---------------- LayerNorm + FiLM: xc = ln(x)*(g+1)+b -----------------------
__global__ __launch_bounds__(256) void k_lnfilm(const float* __restrict__ X,
                                                const float* __restrict__ fbase,
                                                float* __restrict__ XC) {
  __shared__ float red[256];
  const int row = blockIdx.x;
  const int b = row >> 10;
  const float* x = X + (long)row * DIMC;
  const float* g = fbase + (long)b * 6144;
  const float* bb = g + 1024;
  float s = 0.f;
  for (int d = threadIdx.x; d < DIMC; d += 256) s += x[d];
  red[threadIdx.x] = s;
  __syncthreads();
  for (int st = 128; st > 0; st >>= 1) {
    if (threadIdx.x < st) red[threadIdx.x] += red[threadIdx.x + st];
    __syncthreads();
  }
  float mu = red[0] * (1.0f / DIMC);
  __syncthreads();
  float v = 0.f;
  for (int d = threadIdx.x; d < DIMC; d += 256) {
    float t = x[d] - mu;
    v += t * t;
  }
  red[threadIdx.x] = v;
  __syncthreads();
  for (int st = 128; st > 0; st >>= 1) {
    if (threadIdx.x < st) red[threadIdx.x] += red[threadIdx.x + st];
    __syncthreads();
  }
  float rstd = rsqrtf(red[0] * (1.0f / DIMC) + 1e-5f);
  __syncthreads();
  for (int d = threadIdx.x; d < DIMC; d += 256) {
    float xn = (x[d] - mu) * rstd;
    XC[(long)row * DIMC + d] = xn * (g[d] + 1.0f) + bb[d];
  }
}

// ---------------- head gates: sigmoid(xc @ gates_w) --------------------------
__global__ __launch_bounds__(128) void k_gates(const float* __restrict__ XC,
                                               const float* __restrict__ gw,
                                               float* __restrict__ gates) {
  __shared__ float red[128];
  const int row = blockIdx.x;
  const int j = threadIdx.x & 15;
  const int seg = threadIdx.x >> 4;  // 0..7
  const float* x = XC + (long)row * DIMC;
  float acc = 0.f;
  for (int k = seg; k < DIMC; k += 8) acc += x[k] * gw[k * NHEAD + j];
  red[threadIdx.x] = acc;
  __syncthreads();
  for (int st = 4; st > 0; st >>= 1) {
    if (seg < st) red[seg * 16 + j] += red[(seg + st) * 16 + j];
    __syncthreads();
  }
  if (seg == 0) gates[row * NHEAD + j] = sigmoidf_(red[j]);
}

// ---------------- qkv post: rope(q)*scale, rope(k), v mixing -----------------
__global__ __launch_bounds__(64) void k_qkvpost(
    const float* __restrict__ qkvraw, const float* __restrict__ rfreq,
    float* __restrict__ Q, float* __restrict__ Ko, float* __restrict__ Vo,
    float* __restrict__ VR, int layer) {
  __shared__ float qs[64], ks[64];
  const int d = threadIdx.x;
  const int n = blockIdx.x, h = blockIdx.y, b = blockIdx.z;
  const long row = (long)(b * NSEQ + n) * (3 * NHEAD * DH);
  float qv = qkvraw[row + h * DH + d];
  float kv = qkvraw[row + NHEAD * DH + h * DH + d];
  float vv = qkvraw[row + 2 * NHEAD * DH + h * DH + d];
  qs[d] = qv;
  ks[d] = kv;
  __syncthreads();
  float rf = rfreq[n * DH + d];
  float c = cosf(rf), s = sinf(rf);
  int partner = d ^ 1;
  float sign = (d & 1) ? 1.0f : -1.0f;  // rotate_half: even -> -t[d+1], odd -> +t[d-1]
  float qr = (qv * c + qs[partner] * sign * s) * 0.125f;  // * DIM_HEAD^-0.5
  float kr = kv * c + ks[partner] * sign * s;
  const long ob = (((long)(b * NHEAD + h)) * NSEQ + n) * DH + d;
  Q[ob] = qr;
  Ko[ob] = kr;
  if (layer == 0) {
    Vo[ob] = vv;
    VR[ob] = vv;
  } else {
    Vo[ob] = 0.5f * (vv + VR[ob]);
  }
}

// ---------------- flash attention with softcap + causal, WMMA fp32 ----------
#define FSTR 68  // LDS row stride: conflict-free ((4*lr+k)%64) and 16B-aligned
__global__ __launch_bounds__(32) void k_flash(const float* __restrict__ Q,
                                              const float* __restrict__ Kp,
                                              const float* __restrict__ Vp,
                                              float* __restrict__ O) {
  __shared__ float Qs[16][FSTR];
  __shared__ float Ks[16][FSTR];
  __shared__ float Vs[16][FSTR];
  __shared__ float Ps[16][17];
  const int lane = threadIdx.x;
  const int hf = lane >> 4;
  const int lr = lane & 15;
  const int qt = blockIdx.x;   // query tile 0..63
  const int h = blockIdx.y;
  const int b = blockIdx.z;
  const long base = ((long)(b * NHEAD + h)) * NSEQ * DH;
  const float* qp = Q + base;
  const float* kp = Kp + base;
  const float* vp = Vp + base;

  // async-load Q tile (16x64): 256 b128 segments / 32 lanes = 8 per lane
#pragma unroll
  for (int it = 0; it < 8; ++it) {
    int seg = lane + 32 * it;
    int r = seg >> 4, c4 = (seg & 15) * 4;
    async_b128(lds_addr(&Qs[r][c4]), qp + (long)(qt * 16 + r) * DH + c4);
  }
  wait_async0();
  __syncthreads();

  float mrow[8], lrow[8];
  v8f oacc[4];
#pragma unroll
  for (int r = 0; r < 8; ++r) { mrow[r] = NEGMAX; lrow[r] = 0.0f; }
#pragma unroll
  for (int t = 0; t < 4; ++t) oacc[t] = (v8f){0.f,0.f,0.f,0.f,0.f,0.f,0.f,0.f};

  for (int j0 = 0; j0 <= qt; ++j0) {
#pragma unroll
    for (int it = 0; it < 8; ++it) {
      int seg = lane + 32 * it;
      int r = seg >> 4, c4 = (seg & 15) * 4;
      async_b128(lds_addr(&Ks[r][c4]), kp + (long)(j0 * 16 + r) * DH + c4);
      async_b128(lds_addr(&Vs[r][c4]), vp + (long)(j0 * 16 + r) * DH + c4);
    }
    wait_async0();
    __syncthreads();
    v8f sacc = {0.f,0.f,0.f,0.f,0.f,0.f,0.f,0.f};
#pragma unroll
    for (int k4 = 0; k4 < 64; k4 += 4) {
      int ksx = k4 + hf * 2;
      v2f a = {Qs[lr][ksx], Qs[lr][ksx + 1]};
      v2f bf = {Ks[lr][ksx], Ks[lr][ksx + 1]};  // B[k][col] = K[col][k]
      sacc = wmma4(a, bf, sacc);
    }
#pragma unroll
    for (int r = 0; r < 8; ++r) {
      float s = 50.0f * tanhf(sacc[r] * 0.02f);  // softcap
      int qrow = qt * 16 + hf * 8 + r;
      int kcol = j0 * 16 + lr;
      if (kcol > qrow) s = NEGMAX;  // causal
      float mx = s;
      for (int msk = 1; msk < 16; msk <<= 1)
        mx = fmaxf(mx, __shfl_xor(mx, msk, 16));
      float nm = fmaxf(mrow[r], mx);
      float alpha = expf(mrow[r] - nm);
      float pe = expf(s - nm);
      float rs = pe;
      for (int msk = 1; msk < 16; msk <<= 1) rs += __shfl_xor(rs, msk, 16);
      lrow[r] = lrow[r] * alpha + rs;
      mrow[r] = nm;
      oacc[0][r] *= alpha;
      oacc[1][r] *= alpha;
      oacc[2][r] *= alpha;
      oacc[3][r] *= alpha;
      Ps[hf * 8 + r][lr] = pe;
    }
    __syncthreads();
#pragma unroll
    for (int n0 = 0; n0 < 4; ++n0) {
#pragma unroll
      for (int k4 = 0; k4 < 16; k4 += 4) {
        int ksx = k4 + hf * 2;
        v2f a = {Ps[lr][ksx], Ps[lr][ksx + 1]};
        v2f bf = {Vs[ksx][n0 * 16 + lr], Vs[ksx + 1][n0 * 16 + lr]};
        oacc[n0] = wmma4(a, bf, oacc[n0]);
      }
    }
    __syncthreads();
  }
#pragma unroll
  for (int n0 = 0; n0 < 4; ++n0)
#pragma unroll
    for (int r = 0; r < 8; ++r) {
      int row = qt * 16 + hf * 8 + r;
      O[base + (long)row * DH + n0 * 16 + lr] = oacc[n0][r] / lrow[r];
    }
}

// ---------------- merge heads with gates -------------------------------------
__global__ __launch_bounds__(256) void k_omerge(const float* __restrict__ O,
                                                const float* __restrict__ gates,
                                                float* __restrict__ OM) {
  const int row = blockIdx.x;
  const int b = row >> 10, n = row & 1023;
  for (int c = threadIdx.x; c < DIMC; c += 256) {
    int h = c >> 6, d = c & 63;
    OM[(long)row * DIMC + c] =
        O[(((long)(b * NHEAD + h)) * NSEQ + n) * DH + d] * gates[row * NHEAD + h];
  }
}

// ---------------- x += tmp * adaz (per-batch channel gate) -------------------
__global__ __launch_bounds__(256) void k_addscaled(float* __restrict__ X,
                                                   const float* __restrict__ T,
                                                   const float* __restrict__ az) {
  const int row = blockIdx.x;
  const int b = row >> 10;
  for (int d = threadIdx.x; d < DIMC; d += 256)
    X[(long)row * DIMC + d] += T[(long)row * DIMC + d] * az[(long)b * 6144 + d];
}

// ---------------- concat [x, skip] -------------------------------------------
__global__ __launch_bounds__(256) void k_concat(const float* __restrict__ X,
                                                const float* __restrict__ S,
                                                float* __restrict__ Cat) {
  const int row = blockIdx.x;
  for (int d = threadIdx.x; d < DIMC; d += 256) {
    Cat[(long)row * 2048 + d] = X[(long)row * DIMC + d];
    Cat[(long)row * 2048 + DIMC + d] = S[(long)row * DIMC + d];
  }
}

// ---------------- GEGLU: gg = gelu(h[:,DFF:]) * h[:,:DFF] --------------------
__global__ __launch_bounds__(256) void k_geglu(const float* __restrict__ H,
                                               float* __restrict__ GG) {
  const int row = blockIdx.x;
  for (int j = threadIdx.x; j < DFFC; j += 256) {
    float a = H[(long)row * (2 * DFFC) + j];
    float g = H[(long)row * (2 * DFFC) + DFFC + j];
    float ge = 0.5f * g * (1.0f + erff(g * 0.7071067811865475f));
    GG[(long)row * DFFC + j] = ge * a;
  }
}

// ---------------- final norm: x/||x|| * sqrt(D) * (gamma+1) ------------------
__global__ __launch_bounds__(256) void k_finalnorm(float* __restrict__ X,
                                                   const float* __restrict__ gam) {
  __shared__ float red[256];
  const int row = blockIdx.x;
  float s = 0.f;
  for (int d = threadIdx.x; d < DIMC; d += 256) {
    float t = X[(long)row * DIMC + d];
    s += t * t;
  }
  red[threadIdx.x] = s;
  __syncthreads();
  for (int st = 128; st > 0; st >>= 1) {
    if (threadIdx.x < st) red[threadIdx.x] += red[threadIdx.x + st];
    __syncthreads();
  }
  float sc = 32.0f / fmaxf(sqrtf(red[0]), 1e-12f);  // sqrt(1024) = 32
  __syncthreads();
  for (int d = threadIdx.x; d < DIMC; d += 256)
    X[(long)row * DIMC + d] = X[(long)row * DIMC + d] * sc * (gam[d] + 1.0f);
}

// =============================================================================
extern "C" void kernel_launch(void* const* d_in, const int* in_sizes, int n_in,
                              void* d_out, int out_size, void* d_ws,
                              size_t ws_size, hipStream_t stream) {
  (void)in_sizes; (void)n_in; (void)out_size; (void)ws_size;
  const float* in_x     = (const float*)d_in[0];
  const float* times    = (const float*)d_in[1];
  const float* rfreq    = (const float*)d_in[2];
  const float* fw       = (const float*)d_in[3];
  const float* tcond_w  = (const float*)d_in[4];
  const float* tcond_b  = (const float*)d_in[5];
  const float* skip_w   = (const float*)d_in[6];
  const float* afw      = (const float*)d_in[7];
  const float* afb      = (const float*)d_in[8];
  const float* azw      = (const float*)d_in[9];
  const float* azb      = (const float*)d_in[10];
  const float* qkv_w    = (const float*)d_in[11];
  const float* gates_w  = (const float*)d_in[12];
  const float* out_w    = (const float*)d_in[13];
  const float* ffw      = (const float*)d_in[14];
  const float* ffb      = (const float*)d_in[15];
  const float* fzw      = (const float*)d_in[16];
  const float* fzb      = (const float*)d_in[17];
  const float* ff_w1    = (const float*)d_in[18];
  const float* ff_b1    = (const float*)d_in[19];
  const float* ff_w2    = (const float*)d_in[20];
  const float* ff_b2    = (const float*)d_in[21];
  const float* gamma    = (const float*)d_in[22];

  float* x = (float*)d_out;  // live activation buffer (2048 x 1024)
  float* ws = (float*)d_ws;
  size_t o = 0;
  float* cond   = ws + o; o += (size_t)BSZ * DCOND;             // 8192
  float* films  = ws + o; o += (size_t)NLAYER * BSZ * 6144;     // 73728
  float* gates  = ws + o; o += (size_t)MROWS * NHEAD;           // 32768
  float* xc     = ws + o; o += (size_t)MROWS * DIMC;            // xc / omerged
  float* qb     = ws + o; o += (size_t)MROWS * DIMC;
  float* kb     = ws + o; o += (size_t)MROWS * DIMC;
  float* vb     = ws + o; o += (size_t)MROWS * DIMC;
  float* vr     = ws + o; o += (size_t)MROWS * DIMC;
  float* obuf   = ws + o; o += (size_t)MROWS * DIMC;
  float* tmp    = ws + o; o += (size_t)MROWS * DIMC;
  float* skip   = ws + o; o += (size_t)3 * MROWS * DIMC;
  float* catbuf = ws + o; o += (size_t)MROWS * 2048;
  float* qkvraw = ws + o; o += (size_t)MROWS * 3072;            // also GEGLU out
  float* hbuf   = ws + o; o += (size_t)MROWS * 2 * DFFC;
  float* ggbuf  = qkvraw;  // reuse

  const size_t xbytes = (size_t)MROWS * DIMC * sizeof(float);

  k_cond<<<dim3(16, BSZ), 256, 0, stream>>>(times, fw, tcond_w, tcond_b, cond);
  k_films<<<dim3(24, BSZ, NLAYER), 256, 0, stream>>>(
      cond, afw, afb, azw, azb, ffw, ffb, fzw, fzb, films);
  hipMemcpyAsync(x, in_x, xbytes, hipMemcpyDeviceToDevice, stream);

  for (int l = 0; l < NLAYER; ++l) {
    const long fbase = (long)l * BSZ * 6144;
    if (l < 3) {
      hipMemcpyAsync(skip + (size_t)l * MROWS * DIMC, x, xbytes,
                     hipMemcpyDeviceToDevice, stream);
    } else {
      k_concat<<<MROWS, 256, 0, stream>>>(
          x, skip + (size_t)(NLAYER - 1 - l) * MROWS * DIMC, catbuf);
      k_gemm<<<dim3(1024 / 64, MROWS / 64), 256, 0, stream>>>(
          catbuf, skip_w + (size_t)(l - 3) * 2048 * 1024, nullptr, x,
          MROWS, 2048, 1024);
    }
    // ---- attention block ----
    k_lnfilm<<<MROWS, 256, 0, stream>>>(x, films + fbase, xc);
    k_gates<<<MROWS, 128, 0, stream>>>(xc, gates_w + (size_t)l * DIMC * NHEAD,
                                       gates);
    k_gemm<<<dim3(3072 / 64, MROWS / 64), 256, 0, stream>>>(
        xc, qkv_w + (size_t)l * DIMC * 3072, nullptr, qkvraw, MROWS, DIMC, 3072);
    k_qkvpost<<<dim3(NSEQ, NHEAD, BSZ), 64, 0, stream>>>(qkvraw, rfreq, qb, kb,
                                                         vb, vr, l);
    k_flash<<<dim3(NSEQ / 16, NHEAD, BSZ), 32, 0, stream>>>(qb, kb, vb, obuf);
    k_omerge<<<MROWS, 256, 0, stream>>>(obuf, gates, xc);  // xc = merged o
    k_gemm<<<dim3(1024 / 64, MROWS / 64), 256, 0, stream>>>(
        xc, out_w + (size_t)l * DIMC * DIMC, nullptr, tmp, MROWS, DIMC, DIMC);
    k_addscaled<<<MROWS, 256, 0, stream>>>(x, tmp, films + fbase + 2048);
    // ---- feed-forward block ----
    k_lnfilm<<<MROWS, 256, 0, stream>>>(x, films + fbase + 3072, xc);
    k_gemm<<<dim3((2 * DFFC + 63) / 64, MROWS / 64), 256, 0, stream>>>(
        xc, ff_w1 + (size_t)l * DIMC * 2 * DFFC, ff_b1 + (size_t)l * 2 * DFFC,
        hbuf, MROWS, DIMC, 2 * DFFC);
    k_geglu<<<MROWS, 256, 0, stream>>>(hbuf, ggbuf);
    k_gemm<<<dim3(1024 / 64, MROWS / 64), 256, 0, stream>>>(
        ggbuf, ff_w2 + (size_t)l * DFFC * DIMC, ff_b2 + (size_t)l * DIMC, tmp,
        MROWS, DFFC, DIMC);
    k_addscaled<<<MROWS, 256, 0, stream>>>(x, tmp, films + fbase + 5120);
  }
  k_finalnorm<<<MROWS, 256, 0, stream>>>(x, gamma);
}